// ScatterSplitReplacementModel_83150566851282
// MI455X (gfx1250) — compile-verified
//
#include <hip/hip_runtime.h>

#define NUM_HEADS   32
#define NUM_KV      8
#define HEAD        128
#define HALF        64
#define ROW         6144          // 48 heads * 128 floats
#define Q_SIZE      (NUM_HEADS * HEAD)   // 4096
#define KV_SIZE     (NUM_KV * HEAD)      // 1024
#define WAVES       8
#define BLOCK       256
#define CHUNKS_PER_WAVE 6

typedef float f4 __attribute__((ext_vector_type(4)));

// Raw LDS byte offset of a __shared__ object (addrspacecast generic->local;
// LDS pointers are 32-bit offsets from LDS_BASE).
__device__ __forceinline__ unsigned lds_offset(const void* p) {
    return (unsigned)(size_t)(__attribute__((address_space(3))) const void*)p;
}

// Wait until at most N async (global<->LDS) ops are outstanding for this wave.
// Async loads complete in issue order, so ASYNCcnt <= 5-i means chunks 0..i
// have fully landed in LDS. Literal immediates so the SOPP always assembles.
__device__ __forceinline__ void wait_async_le(int n) {
    switch (n) {
    case 0: asm volatile("s_wait_asynccnt 0" ::: "memory"); break;
    case 1: asm volatile("s_wait_asynccnt 1" ::: "memory"); break;
    case 2: asm volatile("s_wait_asynccnt 2" ::: "memory"); break;
    case 3: asm volatile("s_wait_asynccnt 3" ::: "memory"); break;
    case 4: asm volatile("s_wait_asynccnt 4" ::: "memory"); break;
    default: asm volatile("s_wait_asynccnt 5" ::: "memory"); break;
    }
}

__global__ __launch_bounds__(BLOCK)
void rope_split_qkv_kernel(const float* __restrict__ qkv,
                           const int*   __restrict__ positions,
                           float* __restrict__ out,
                           int ntok)
{
    __shared__ __align__(16) float stage[ROW];   // one token's qkv row, 24 KB
    __shared__ float cs_c[HALF];
    __shared__ float cs_s[HALF];

    const int t    = blockIdx.x;
    const int tid  = threadIdx.x;
    const int wave = tid >> 5;
    const int lane = tid & 31;

    const float*   rowbase    = qkv + (size_t)t * ROW;
    const unsigned stage_base = lds_offset(stage);

    // ---- Stage 1: kick off async DMA of this wave's 6 chunks into LDS ----
    // GVS addressing: mem = SGPR64(rowbase) + VGPR32(byte offset)
#pragma unroll
    for (int i = 0; i < CHUNKS_PER_WAVE; ++i) {
        const int      chunk = wave + WAVES * i;
        const unsigned voff  = (unsigned)(chunk * (HEAD * 4) + lane * 16);
        const unsigned laddr = stage_base + voff;
        asm volatile("global_load_async_to_lds_b128 %0, %1, %2"
                     :
                     : "v"(laddr), "v"(voff), "s"(rowbase)
                     : "memory");
    }

    // ---- Stage 2 (overlaps the DMA): per-token cos/sin table ----
    // Spread the 64 transcendental pairs over all 8 waves (8 entries each)
    // so no wave sits idle at the barrier behind two table-building waves.
    if (lane < 8) {
        const int   j   = wave * 8 + lane;
        const float pos = (float)positions[t];
        // inv_freq[j] = 10000^(-j/64) = exp2(-j * log2(10000)/64)
        const float ang = pos * exp2f((float)j * -0.20762050593046014f);
        cs_c[j] = cosf(ang);
        cs_s[j] = sinf(ang);
    }
    __syncthreads();          // publish cos/sin table to all waves

    // ---- Stage 3: transform + scatter-store (progressive async waits) ----
    float* const oq = out;
    float* const ok = out + (size_t)ntok * Q_SIZE;
    float* const ov = ok  + (size_t)ntok * KV_SIZE;

    const int   jbase = (lane & 15) * 4;          // freq index of first elem
    const float sgn   = (lane < 16) ? -1.0f : 1.0f;

#pragma unroll
    for (int i = 0; i < CHUNKS_PER_WAVE; ++i) {
        wait_async_le(CHUNKS_PER_WAVE - 1 - i);   // chunk i landed in LDS

        const int chunk = wave + WAVES * i;       // i<=3: q, i==4: k, i==5: v
        const f4  x = *(const f4*)&stage[chunk * HEAD + lane * 4];
        f4 r;
        if (i <= 4) {
            // RoPE: partner element (d +/- 64) lives at lane^16 in same chunk
            const f4    p   = *(const f4*)&stage[chunk * HEAD + (lane ^ 16) * 4];
            const float add = (i <= 3) ? 1.0f : 2.0f;
#pragma unroll
            for (int c = 0; c < 4; ++c)
                r[c] = x[c] * cs_c[jbase + c] + sgn * p[c] * cs_s[jbase + c] + add;
        } else {
#pragma unroll
            for (int c = 0; c < 4; ++c)
                r[c] = x[c] + 3.0f;
        }

        float* dst;
        if (i <= 3) {
            dst = oq + (size_t)t * Q_SIZE  + chunk * HEAD + lane * 4;
        } else if (i == 4) {
            dst = ok + (size_t)t * KV_SIZE + (chunk - NUM_HEADS) * HEAD + lane * 4;
        } else {
            dst = ov + (size_t)t * KV_SIZE + (chunk - NUM_HEADS - NUM_KV) * HEAD + lane * 4;
        }
        __builtin_nontemporal_store(r, (f4*)dst);   // write-once stream: NT
    }
}

extern "C" void kernel_launch(void* const* d_in, const int* in_sizes, int n_in,
                              void* d_out, int out_size, void* d_ws, size_t ws_size,
                              hipStream_t stream) {
    const float* qkv       = (const float*)d_in[0];
    const int*   positions = (const int*)d_in[1];
    float*       out       = (float*)d_out;
    const int    ntok      = in_sizes[0] / ROW;

    rope_split_qkv_kernel<<<ntok, BLOCK, 0, stream>>>(qkv, positions, out, ntok);
}